// GAT_188978561447
// MI455X (gfx1250) — compile-verified
//
#include <hip/hip_runtime.h>
#include <math.h>

// ---------------------------------------------------------------------------
// 2-layer GAT for MI455X (gfx1250, wave32).
//  - GEMMs via V_WMMA_F32_16X16X4_F32 (full fp32 precision, matches reference)
//  - segment softmax via monotone-uint atomic max + f32 atomic adds (L2 resident)
//  - per-edge alpha precomputed once; aggregation vectorized float4 + atomics
// ---------------------------------------------------------------------------

typedef __attribute__((ext_vector_type(2))) float v2f;
typedef __attribute__((ext_vector_type(8))) float v8f;

#define NEG_SLOPE 0.2f

#if defined(__has_builtin)
#if __has_builtin(__builtin_amdgcn_wmma_f32_16x16x4_f32)
#define HAVE_WMMA_F32_16X16X4 1
#endif
#endif

// ---- monotone mapping of float -> uint so unsigned atomicMax == float max ----
__device__ __forceinline__ unsigned flipf(float f) {
  unsigned b = __float_as_uint(f);
  return (b & 0x80000000u) ? ~b : (b | 0x80000000u);
}
__device__ __forceinline__ float unflipf(unsigned u) {
  return __uint_as_float((u & 0x80000000u) ? (u & 0x7FFFFFFFu) : ~u);
}

__device__ __forceinline__ float leaky(float v) {
  return v > 0.f ? v : NEG_SLOPE * v;
}

// ---------------------------------------------------------------------------
// H[M,DOUT] = act(X)[M,K] @ W[K,DOUT]
// act = identity, or relu(x + bias[k]) when RELU_BIAS (fuses layer-1 epilogue
// into the layer-2 GEMM load).
// One wave32 per 16x16 output tile; K consumed 4 at a time with
// v_wmma_f32_16x16x4_f32.
//
// fp32 A 16x4 per-lane layout: lanes 0-15 hold rows 0-15 with K = {k0, k0+1}
// in (v0,v1); lanes 16-31 hold the same rows with K = {k0+2, k0+3}.
// fp32 B 4x16 mirrored (row = K, striped across lanes).
// C/D 16x16: vgpr r -> row r (lanes 0-15) / row r+8 (lanes 16-31), col = lane%16.
// ---------------------------------------------------------------------------
template <int K, int DOUT, bool RELU_BIAS>
__global__ void gat_gemm_wmma(const float* __restrict__ X,
                              const float* __restrict__ W,
                              const float* __restrict__ bias,
                              float* __restrict__ H, int M) {
  const int lane   = threadIdx.x & 31;
  const int wave   = threadIdx.x >> 5;
  const int tileId = blockIdx.x * (blockDim.x >> 5) + wave;
  const int NT     = DOUT / 16;
  const int mt     = tileId / NT;
  const int nt     = tileId % NT;
  if (mt >= M / 16) return;  // wave-uniform exit: EXEC stays all-ones for WMMA

  const int r15 = lane & 15;
  const int hi  = lane >> 4;  // 0: K pair {0,1}, 1: K pair {2,3}
  const int row = mt * 16 + r15;
  const int col = nt * 16 + r15;

  v8f c = {0.f, 0.f, 0.f, 0.f, 0.f, 0.f, 0.f, 0.f};

#if HAVE_WMMA_F32_16X16X4
  for (int k0 = 0; k0 < K; k0 += 4) {
    const int ka = k0 + 2 * hi;
    float x0 = X[row * K + ka];
    float x1 = X[row * K + ka + 1];
    if (RELU_BIAS) {
      x0 = fmaxf(x0 + bias[ka], 0.f);
      x1 = fmaxf(x1 + bias[ka + 1], 0.f);
    }
    v2f a;
    a.x = x0;
    a.y = x1;
    v2f b;
    b.x = W[ka * DOUT + col];
    b.y = W[(ka + 1) * DOUT + col];
    // (neg_a, A, neg_b, B, c_mod, C, reuse_a, reuse_b)
    c = __builtin_amdgcn_wmma_f32_16x16x4_f32(false, a, false, b, (short)0, c,
                                              false, false);
  }
#else
  // Scalar fallback (keeps compile alive if the fp32 WMMA builtin differs).
  for (int r = 0; r < 8; ++r) {
    const int rr = mt * 16 + hi * 8 + r;
    float acc = 0.f;
    for (int k = 0; k < K; ++k) {
      float xv = X[rr * K + k];
      if (RELU_BIAS) xv = fmaxf(xv + bias[k], 0.f);
      acc += xv * W[k * DOUT + col];
    }
    c[r] = acc;
  }
#endif

  const int rowBase = mt * 16 + hi * 8;
#pragma unroll
  for (int r = 0; r < 8; ++r) H[(rowBase + r) * DOUT + col] = c[r];
}

// ---------------------------------------------------------------------------
// Per-node attention scores: aS[i] = dot(H[i], attS), aD[i] = dot(H[i], attD).
// One wave per node, cross-lane reduce (wave32 shuffles).
// ---------------------------------------------------------------------------
template <int D>
__global__ void att_scores(const float* __restrict__ H,
                           const float* __restrict__ attS,
                           const float* __restrict__ attD,
                           float* __restrict__ aS, float* __restrict__ aD,
                           int N) {
  const int gid  = blockIdx.x * blockDim.x + threadIdx.x;
  const int node = gid >> 5;
  const int lane = gid & 31;
  if (node >= N) return;
  float s = 0.f, d = 0.f;
#pragma unroll
  for (int f = lane; f < D; f += 32) {
    const float h = H[(size_t)node * D + f];
    s += h * attS[f];
    d += h * attD[f];
  }
#pragma unroll
  for (int off = 16; off > 0; off >>= 1) {
    s += __shfl_down(s, off, 32);
    d += __shfl_down(d, off, 32);
  }
  if (lane == 0) {
    aS[node] = s;
    aD[node] = d;
  }
}

// ---------------------------------------------------------------------------
// Edge pass 1: e = leaky_relu(aS[src] + aD[dst]); segment max into emax[dst].
// Edges e >= E are the appended self-loops (src = dst = e - E).
// ---------------------------------------------------------------------------
__global__ void edge_lrelu_max(const int* __restrict__ src,
                               const int* __restrict__ dst,
                               const float* __restrict__ aS,
                               const float* __restrict__ aD,
                               unsigned* __restrict__ emax, int E, int Etot) {
  const int e = blockIdx.x * blockDim.x + threadIdx.x;
  if (e >= Etot) return;
  int s, d;
  if (e < E) {
    s = src[e];
    d = dst[e];
  } else {
    s = d = e - E;
  }
  const float v = leaky(aS[s] + aD[d]);
  atomicMax(&emax[d], flipf(v));  // global_atomic_max_u32
}

// ---------------------------------------------------------------------------
// Edge pass 2: w = exp(e - emax[dst]); wbuf[e] = w; segment sum into denom.
// ---------------------------------------------------------------------------
__global__ void edge_expsum(const int* __restrict__ src,
                            const int* __restrict__ dst,
                            const float* __restrict__ aS,
                            const float* __restrict__ aD,
                            const unsigned* __restrict__ emax,
                            float* __restrict__ wbuf,
                            float* __restrict__ denom, int E, int Etot) {
  const int e = blockIdx.x * blockDim.x + threadIdx.x;
  if (e >= Etot) return;
  int s, d;
  if (e < E) {
    s = src[e];
    d = dst[e];
  } else {
    s = d = e - E;
  }
  const float v = leaky(aS[s] + aD[d]);
  const float m = unflipf(emax[d]);
  const float w = __expf(v - m);
  wbuf[e] = w;
  unsafeAtomicAdd(&denom[d], w);  // global_atomic_add_f32
}

// ---------------------------------------------------------------------------
// Edge pass 2.5: normalize once per edge: wbuf[e] /= denom[dst[e]].
// Moves the f32 divide out of the per-(edge,feature) hot loop (1.7M divides
// instead of 109M).
// ---------------------------------------------------------------------------
__global__ void edge_alpha(const int* __restrict__ dst,
                           const float* __restrict__ denom,
                           float* __restrict__ wbuf, int E, int Etot) {
  const int e = blockIdx.x * blockDim.x + threadIdx.x;
  if (e >= Etot) return;
  const int d = (e < E) ? dst[e] : (e - E);
  wbuf[e] = wbuf[e] / denom[d];
}

// ---------------------------------------------------------------------------
// Edge pass 3: out[dst] += alpha[e] * H[src]; one thread per (edge, 4 feats).
// float4 gather (global_load_b128) of the H row + 4 f32 atomics.
// Gathers hit L2 (H is 25.6 MB << 192 MB L2).
// ---------------------------------------------------------------------------
template <int D>
__global__ void edge_aggregate(const int* __restrict__ src,
                               const int* __restrict__ dst,
                               const float* __restrict__ alpha,
                               const float* __restrict__ H,
                               float* __restrict__ out, int E, int Etot) {
  constexpr int TPE = D / 4;  // threads per edge (power of 2)
  const long long t = (long long)blockIdx.x * blockDim.x + threadIdx.x;
  const long long e = t / TPE;
  const int q = (int)(t % TPE);  // feature quad
  if (e >= Etot) return;
  int s, d;
  if (e < E) {
    s = src[e];
    d = dst[e];
  } else {
    s = d = (int)(e - E);
  }
  const float a = alpha[e];
  const float4 h = *reinterpret_cast<const float4*>(H + (size_t)s * D + 4 * q);
  float* o = out + (size_t)d * D + 4 * q;
  unsafeAtomicAdd(o + 0, a * h.x);
  unsafeAtomicAdd(o + 1, a * h.y);
  unsafeAtomicAdd(o + 2, a * h.z);
  unsafeAtomicAdd(o + 3, a * h.w);
}

// ---------------------------------------------------------------------------
// Final epilogue: out = tanh(out + b2[f]).
// ---------------------------------------------------------------------------
template <int D>
__global__ void finalize_tanh(float* __restrict__ out,
                              const float* __restrict__ b, int N) {
  const int t = blockIdx.x * blockDim.x + threadIdx.x;
  if (t >= N * D) return;
  out[t] = tanhf(out[t] + b[t & (D - 1)]);
}

// ---------------------------------------------------------------------------
extern "C" void kernel_launch(void* const* d_in, const int* in_sizes, int n_in,
                              void* d_out, int out_size, void* d_ws,
                              size_t ws_size, hipStream_t stream) {
  const float* x   = (const float*)d_in[0];
  const int*   ei  = (const int*)d_in[1];
  const float* W1  = (const float*)d_in[2];
  const float* as1 = (const float*)d_in[3];
  const float* ad1 = (const float*)d_in[4];
  const float* b1  = (const float*)d_in[5];
  const float* W2  = (const float*)d_in[6];
  const float* as2 = (const float*)d_in[7];
  const float* ad2 = (const float*)d_in[8];
  const float* b2  = (const float*)d_in[9];
  float* out = (float*)d_out;

  const int N    = in_sizes[0] / 64;
  const int E    = in_sizes[1] / 2;
  const int Etot = E + N;
  const int* src = ei;      // edge_index[0]
  const int* dst = ei + E;  // edge_index[1]

  // ---- workspace layout (floats; all offsets 16B-aligned) ----
  float* ws = (float*)d_ws;
  size_t o = 0;
  float* h1 = ws + o;   o += (size_t)N * 64;  // layer-1 h; reused as layer-2 h
  float* o1 = ws + o;   o += (size_t)N * 64;  // layer-1 aggregated output
  float* aS = ws + o;   o += N;
  float* aD = ws + o;   o += N;
  unsigned* emax = (unsigned*)(ws + o); o += N;
  float* denom = ws + o; o += N;
  float* wbuf  = ws + o; o += Etot;           // edge weights -> alpha in place

  const int B  = 256;
  const int ge = (Etot + B - 1) / B;

  // ================= layer 1 =================
  {
    const int tiles = (N / 16) * (64 / 16);
    gat_gemm_wmma<64, 64, false>
        <<<(tiles + 7) / 8, B, 0, stream>>>(x, W1, nullptr, h1, N);
    att_scores<64><<<(int)(((size_t)N * 32 + B - 1) / B), B, 0, stream>>>(
        h1, as1, ad1, aS, aD, N);
    hipMemsetAsync(emax, 0, (size_t)N * 4, stream);
    hipMemsetAsync(denom, 0, (size_t)N * 4, stream);
    hipMemsetAsync(o1, 0, (size_t)N * 64 * 4, stream);
    edge_lrelu_max<<<ge, B, 0, stream>>>(src, dst, aS, aD, emax, E, Etot);
    edge_expsum<<<ge, B, 0, stream>>>(src, dst, aS, aD, emax, wbuf, denom, E,
                                      Etot);
    edge_alpha<<<ge, B, 0, stream>>>(dst, denom, wbuf, E, Etot);
    edge_aggregate<64>
        <<<(int)(((size_t)Etot * 16 + B - 1) / B), B, 0, stream>>>(
            src, dst, wbuf, h1, o1, E, Etot);
  }

  // ================= layer 2 =================
  {
    float* h2 = h1;  // safe reuse: stream-ordered, h1 dead after layer-1 agg
    const int tiles = (N / 16) * (32 / 16);
    // relu(o1 + b1) fused into the A-operand load
    gat_gemm_wmma<64, 32, true>
        <<<(tiles + 7) / 8, B, 0, stream>>>(o1, W2, b1, h2, N);
    att_scores<32><<<(int)(((size_t)N * 32 + B - 1) / B), B, 0, stream>>>(
        h2, as2, ad2, aS, aD, N);
    hipMemsetAsync(emax, 0, (size_t)N * 4, stream);
    hipMemsetAsync(denom, 0, (size_t)N * 4, stream);
    hipMemsetAsync(out, 0, (size_t)N * 32 * 4, stream);
    edge_lrelu_max<<<ge, B, 0, stream>>>(src, dst, aS, aD, emax, E, Etot);
    edge_expsum<<<ge, B, 0, stream>>>(src, dst, aS, aD, emax, wbuf, denom, E,
                                      Etot);
    edge_alpha<<<ge, B, 0, stream>>>(dst, denom, wbuf, E, Etot);
    edge_aggregate<32>
        <<<(int)(((size_t)Etot * 8 + B - 1) / B), B, 0, stream>>>(
            src, dst, wbuf, h2, out, E, Etot);
    finalize_tanh<32><<<(int)(((size_t)N * 32 + B - 1) / B), B, 0, stream>>>(
        out, b2, N);
  }
}